// CharModel_86517821212099
// MI455X (gfx1250) — compile-verified
//
#include <hip/hip_runtime.h>

typedef float v2f __attribute__((ext_vector_type(2)));
typedef float v8f __attribute__((ext_vector_type(8)));

namespace {
constexpr int kB = 32;
constexpr int kS = 2048;
constexpr int kW = 512;
constexpr int kD = 768;
}

// One block = one (batch, 16-word tile). 8 wave32s; each wave owns 6 of the
// 48 d-tiles (16 columns each). Segment mean is computed as a block-banded
// GEMM with V_WMMA_F32_16X16X4_F32: A[16w x 4k] holds 1/denom inside each
// word's char range, B[4k x 16d] holds feats rows, C accumulates the means.
// Inner loop: main part unrolled x4 (16 chars / iter, 8 loads in flight,
// immediate-offset addressing), one clamped tail iteration (A==0 past khi
// makes clamped loads mathematically dead).
__global__ __launch_bounds__(256) void word_mean_pos_wmma(
    const float* __restrict__ feats,      // [B, S, D]
    const float* __restrict__ pos_table,  // [PV, D] (row 0 == zeros)
    const int*   __restrict__ word_starts,// [B, W] ("word_lens" == start offsets)
    const int*   __restrict__ pos,        // [B, W]
    const int*   __restrict__ seq_len,    // [B]
    float*       __restrict__ out)        // [B, W, D]
{
    __shared__ int   s_start[16];
    __shared__ int   s_end[16];
    __shared__ float s_inv[16];
    __shared__ int   s_posrow[16];
    __shared__ int   s_klo, s_khi;

    const int blk = blockIdx.x;
    const int b   = blk >> 5;          // batch
    const int w0  = (blk & 31) << 4;   // first word of tile
    const int tid = threadIdx.x;

    // --- per-word metadata (reference semantics) ---
    if (tid < 16) {
        const int w   = w0 + tid;
        const int st  = word_starts[b * kW + w];
        const int nxt = (w + 1 < kW) ? word_starts[b * kW + w + 1] : 0;
        const int sl  = seq_len[b];
        const bool is_last = (w == kW - 1) || (nxt == 0);
        const int  en      = is_last ? sl : nxt;
        const bool valid   = (st != 0) || (w == 0);
        int denom = en - st;
        if (denom < 1) denom = 1;
        // Sentinel start for invalid words => A row is all zeros => mean 0.
        s_start[tid]  = valid ? st : 0x7FFFFFFF;
        s_end[tid]    = valid ? en : (-0x7FFFFFFF);
        s_inv[tid]    = 1.0f / (float)denom;
        s_posrow[tid] = pos[b * kW + w];
    }
    __syncthreads();
    if (tid == 0) {
        int lo = 0x7FFFFFFF, hi = 0;
        #pragma unroll
        for (int m = 0; m < 16; ++m) {
            const int st = s_start[m];
            const int en = s_end[m];
            lo = (st < lo) ? st : lo;
            hi = (en > hi) ? en : hi;
        }
        s_klo = lo;
        s_khi = hi;
    }
    __syncthreads();

    // Scalarize the tile's char span so the k-loop is wave-uniform
    // (WMMA requires EXEC == all ones).
    const int klo = __builtin_amdgcn_readfirstlane(s_klo);
    const int khi = __builtin_amdgcn_readfirstlane(s_khi);

    const int lane = tid & 31;
    const int wave = tid >> 5;
    const int half = lane >> 4;    // 0: K={0,1}, 1: K={2,3} (A and B fragments)
    const int n    = lane & 15;    // A: word row M; B: column N
    const int koff = half << 1;

    // Per-lane A-row constants (M = lane & 15 per the 16x4 A layout).
    const int      astart = s_start[n];
    const unsigned urange = (unsigned)(s_end[n] - astart); // false for sentinel
    const float    ainv   = s_inv[n];

    const float* fcol0 = feats + (size_t)b * kS * kD + n;  // column n, row 0

    for (int dt = wave; dt < kD / 16; dt += 8) {
        const int d0 = dt << 4;
        const float* col = fcol0 + d0;   // &feats[b][0][d0 + n]
        v8f c = {};

        // ---- main loop: 16 chars per iteration, pure linear addressing ----
        int k0 = klo;
        const float* p = col + (size_t)(klo + koff) * kD;
        for (; k0 + 16 <= khi; k0 += 16, p += 16 * (size_t)kD) {
            __builtin_prefetch(p + 16 * (size_t)kD, 0, 3);
            v2f Af[4], Bm[4];
            #pragma unroll
            for (int j = 0; j < 4; ++j) {
                const int ka = k0 + 4 * j + koff;
                Bm[j].x = p[(size_t)(4 * j) * kD];
                Bm[j].y = p[(size_t)(4 * j + 1) * kD];
                Af[j].x = ((unsigned)(ka     - astart) < urange) ? ainv : 0.0f;
                Af[j].y = ((unsigned)(ka + 1 - astart) < urange) ? ainv : 0.0f;
            }
            #pragma unroll
            for (int j = 0; j < 4; ++j)
                c = __builtin_amdgcn_wmma_f32_16x16x4_f32(
                        false, Af[j], false, Bm[j], (short)0, c, false, false);
        }

        // ---- clamped tail: up to 15 remaining chars; overshoot has A==0,
        //      loads clamped to the last in-bounds row (value is dead). ----
        if (k0 < khi) {
            const int khi1 = khi - 1;
            v2f Af[4], Bm[4];
            #pragma unroll
            for (int j = 0; j < 4; ++j) {
                const int ka  = k0 + 4 * j + koff;
                const int kb  = ka + 1;
                const int ksa = (ka < khi1) ? ka : khi1;
                const int ksb = (kb < khi1) ? kb : khi1;
                Bm[j].x = col[(size_t)ksa * kD];
                Bm[j].y = col[(size_t)ksb * kD];
                Af[j].x = ((unsigned)(ka - astart) < urange) ? ainv : 0.0f;
                Af[j].y = ((unsigned)(kb - astart) < urange) ? ainv : 0.0f;
            }
            #pragma unroll
            for (int j = 0; j < 4; ++j)
                c = __builtin_amdgcn_wmma_f32_16x16x4_f32(
                        false, Af[j], false, Bm[j], (short)0, c, false, false);
        }

        // C/D layout: VGPR r -> row M = r + 8*half, column N = lane & 15.
        #pragma unroll
        for (int r = 0; r < 8; ++r) {
            const int    M  = r + (half << 3);
            const size_t po = (size_t)s_posrow[M] * kD + d0 + n;
            const size_t oo = ((size_t)(b * kW + w0 + M)) * (size_t)kD + d0 + n;
            out[oo] = c[r] + pos_table[po];
        }
    }
}

extern "C" void kernel_launch(void* const* d_in, const int* in_sizes, int n_in,
                              void* d_out, int out_size, void* d_ws, size_t ws_size,
                              hipStream_t stream) {
    const float* feats     = (const float*)d_in[0];
    const float* pos_table = (const float*)d_in[1];
    const int*   word_lens = (const int*)d_in[2];
    const int*   pos       = (const int*)d_in[3];
    const int*   seq_len   = (const int*)d_in[4];
    float*       out       = (float*)d_out;

    const dim3 grid(kB * (kW / 16));   // 32 batches x 32 word-tiles = 1024 blocks
    const dim3 block(256);             // 8 wave32s
    word_mean_pos_wmma<<<grid, block, 0, stream>>>(
        feats, pos_table, word_lens, pos, seq_len, out);
}